// TpsGridGen_78245714199406
// MI455X (gfx1250) — compile-verified
//
#include <hip/hip_runtime.h>
#include <hip/hip_bf16.h>

// ---------------------------------------------------------------------------
// TPS grid generator for MI455X (gfx1250).
//  out[b,h,w,c] = sum_{k<28} basis_k(pixel) * coef[c][k][b]
//  basis = [U_0..U_24, 1, gx, gy],  coef = Li[:, :25] @ (theta_c + P_c)
//  Main contraction done with V_WMMA_F32_16X16X4_F32 (exact fp32, K=28=7x4).
// ---------------------------------------------------------------------------

#define OUT_H   256
#define OUT_W   192
#define HW      (OUT_H * OUT_W)      // 49152
#define NB      64                   // batch
#define NCP     25                   // control points
#define NK      28                   // NCP + 3 affine
#define NAUG    56                   // 2*NK augmented columns

typedef __attribute__((ext_vector_type(2))) float v2f;
typedef __attribute__((ext_vector_type(8))) float v8f;

// ---------------------------------------------------------------------------
// Kernel A: build the 28x28 TPS system matrix L in float64 and invert it with
// Gauss-Jordan + partial pivoting (the K-block diagonal is zero, so pivoting
// is required).  Writes Li (row-major 28x28) as float32 into workspace.
// Single wave; cost is negligible.
// ---------------------------------------------------------------------------
__global__ void tps_invert_kernel(float* __restrict__ Li_out) {
    __shared__ double Aug[NK][NAUG];
    __shared__ int pivrow;
    const int tid = threadIdx.x;   // 32 threads

    for (int idx = tid; idx < NK * NAUG; idx += 32) {
        int i = idx / NAUG, j = idx - (idx / NAUG) * NAUG;
        double v;
        if (j >= NK) {
            v = (j - NK == i) ? 1.0 : 0.0;       // identity block
        } else if (i < NCP && j < NCP) {
            double xi = -1.0 + 0.5 * (double)(i / 5);
            double yi = -1.0 + 0.5 * (double)(i % 5);
            double xj = -1.0 + 0.5 * (double)(j / 5);
            double yj = -1.0 + 0.5 * (double)(j % 5);
            double d2 = (xi - xj) * (xi - xj) + (yi - yj) * (yi - yj);
            if (d2 == 0.0) d2 = 1.0;
            v = d2 * log(d2);                    // U kernel
        } else if (i < NCP) {                    // P block: [1, x, y]
            int t = j - NCP;
            v = (t == 0) ? 1.0
              : (t == 1) ? (-1.0 + 0.5 * (double)(i / 5))
                         : (-1.0 + 0.5 * (double)(i % 5));
        } else if (j < NCP) {                    // P^T block
            int t = i - NCP;
            v = (t == 0) ? 1.0
              : (t == 1) ? (-1.0 + 0.5 * (double)(j / 5))
                         : (-1.0 + 0.5 * (double)(j % 5));
        } else {
            v = 0.0;                             // 3x3 zero corner
        }
        Aug[i][j] = v;
    }
    __syncthreads();

    for (int col = 0; col < NK; ++col) {
        if (tid == 0) {
            int best = col;
            double ba = fabs(Aug[col][col]);
            for (int r = col + 1; r < NK; ++r) {
                double a = fabs(Aug[r][col]);
                if (a > ba) { ba = a; best = r; }
            }
            pivrow = best;
        }
        __syncthreads();
        const int pr = pivrow;
        if (pr != col) {
            for (int j = tid; j < NAUG; j += 32) {
                double t = Aug[col][j]; Aug[col][j] = Aug[pr][j]; Aug[pr][j] = t;
            }
        }
        __syncthreads();
        const double piv = Aug[col][col];
        for (int j = tid; j < NAUG; j += 32) Aug[col][j] /= piv;
        __syncthreads();
        if (tid < NK && tid != col) {
            double f = Aug[tid][col];
            for (int j = 0; j < NAUG; ++j) Aug[tid][j] -= f * Aug[col][j];
        }
        __syncthreads();
    }

    for (int idx = tid; idx < NK * NK; idx += 32)
        Li_out[idx] = (float)Aug[idx / NK][NK + idx % NK];
}

// ---------------------------------------------------------------------------
// Kernel B: coefficients.  3584 threads; thread (c,k,b) computes
//   coef[c][k][b] = sum_{m<25} Li[k][m] * (theta[b][c*25+m] + P_c[m])
// stored pair-interleaved for b64 A-fragment loads:
//   index = ((c*14 + (k>>1))*64 + b)*2 + (k&1)
// ---------------------------------------------------------------------------
__global__ void tps_coef_kernel(const float* __restrict__ theta,
                                const float* __restrict__ Li,
                                float* __restrict__ coef) {
    const int t = blockIdx.x * blockDim.x + threadIdx.x;   // 0..3583 exact
    const int c   = t / (NK * NB);
    const int rem = t - c * (NK * NB);
    const int k   = rem >> 6;
    const int b   = rem & 63;

    const float* th = theta + b * (2 * NCP) + c * NCP;
    const float* li = Li + k * NK;
    float sum = 0.0f;
#pragma unroll
    for (int m = 0; m < NCP; ++m) {
        float pc = (c == 0) ? (-1.0f + 0.5f * (float)(m / 5))
                            : (-1.0f + 0.5f * (float)(m % 5));
        sum += li[m] * (th[m] + pc);
    }
    coef[(((c * 14) + (k >> 1)) * NB + b) * 2 + (k & 1)] = sum;
}

// ---------------------------------------------------------------------------
// Kernel C: main pixel kernel.  8 waves / 256 threads per block; each wave
// owns 16 pixels.  Each lane computes its 14 basis values in place per the
// f32 WMMA B-layout (lane -> pixel column, lane-half+vgpr -> K), then runs
// 7 V_WMMA_F32_16X16X4_F32 per coord per 16-batch tile.  Stores are float2
// ({px,py}) -> two contiguous 128B segments per instruction.
// ---------------------------------------------------------------------------
__global__ void __launch_bounds__(256) tps_main_kernel(
        const float* __restrict__ coef, float* __restrict__ out) {
    __shared__ __align__(16) float sC[2 * 14 * NB * 2];    // 3584 floats

    for (int i = threadIdx.x; i < 3584; i += 256) sC[i] = coef[i];
    __syncthreads();

    const int lane = threadIdx.x & 31;
    const int wave = threadIdx.x >> 5;
    const int tile = blockIdx.x * 8 + wave;                // 3072 tiles total
    const int pixbase = tile * 16;
    const int ncol = lane & 15;                            // pixel column / A row
    const int half = lane >> 4;                            // K half select
    const int p = pixbase + ncol;
    const int h = p / OUT_W;
    const int w = p - h * OUT_W;
    const float gx = -1.0f + (2.0f / 191.0f) * (float)w;
    const float gy = -1.0f + (2.0f / 255.0f) * (float)h;

    // ---- B fragments: basis values for this lane's pixel -------------------
    v2f bfrag[7];
#pragma unroll
    for (int kk = 0; kk < 7; ++kk) {
#pragma unroll
        for (int j = 0; j < 2; ++j) {
            const int k = 4 * kk + 2 * half + j;
            const int ki = (k < NCP) ? k : 0;
            const float pxn = -1.0f + 0.5f * (float)(ki / 5);
            const float pyn = -1.0f + 0.5f * (float)(ki % 5);
            const float dx = gx - pxn, dy = gy - pyn;
            float d2 = dx * dx + dy * dy;
            d2 = (d2 == 0.0f) ? 1.0f : d2;
            const float rbf = d2 * __logf(d2);
            // branchless selects keep EXEC all-ones for WMMA
            float u = (k == 25) ? 1.0f
                    : (k == 26) ? gx
                    : (k == 27) ? gy
                                : rbf;
            bfrag[kk][j] = u;
        }
    }

    float2* __restrict__ out2 = (float2*)out;

    for (int bt = 0; bt < 4; ++bt) {                       // 4 batch tiles of 16
        v8f accX = {}, accY = {};
        const int abase = bt * 16 + ncol;                  // A row m = lane&15

#pragma unroll
        for (int kk = 0; kk < 7; ++kk) {
            const int q = 2 * kk + half;                   // pair index of K
            v2f aX = *(const v2f*)&sC[((0 * 14 + q) * NB + abase) * 2];
            accX = __builtin_amdgcn_wmma_f32_16x16x4_f32(
                false, aX, false, bfrag[kk], (short)0, accX, false, false);
        }
#pragma unroll
        for (int kk = 0; kk < 7; ++kk) {
            const int q = 2 * kk + half;
            v2f aY = *(const v2f*)&sC[((1 * 14 + q) * NB + abase) * 2];
            accY = __builtin_amdgcn_wmma_f32_16x16x4_f32(
                false, aY, false, bfrag[kk], (short)0, accY, false, false);
        }

        // D layout: VGPR j -> batch row (j + half*8), lane%16 -> pixel
#pragma unroll
        for (int j = 0; j < 8; ++j) {
            const int b = bt * 16 + j + half * 8;
            float2 v;
            v.x = accX[j];
            v.y = accY[j];
            out2[(size_t)b * HW + p] = v;
        }
    }
}

// ---------------------------------------------------------------------------
extern "C" void kernel_launch(void* const* d_in, const int* in_sizes, int n_in,
                              void* d_out, int out_size, void* d_ws, size_t ws_size,
                              hipStream_t stream) {
    const float* theta = (const float*)d_in[0];   // (64, 50) fp32
    float* out  = (float*)d_out;                  // (64, 256, 192, 2) fp32
    float* Li   = (float*)d_ws;                   // 28*28 fp32
    float* coef = Li + NK * NK;                   // 2*28*64 fp32 (interleaved)

    tps_invert_kernel<<<1, 32, 0, stream>>>(Li);
    tps_coef_kernel<<<14, 256, 0, stream>>>(theta, Li, coef);
    tps_main_kernel<<<(HW / 16) / 8, 256, 0, stream>>>(coef, out);
}